// MultiSegmentLoss_54846732370193
// MI455X (gfx1250) — compile-verified
//
#include <hip/hip_runtime.h>
#include <math.h>

// ---- problem constants (match reference) ----
#define B_DIM  32
#define K_DIM  32768
#define NGT    32
#define NCLS   21
#define CLIPF  256.0f
#define MAXNF  512.0f
#define EPSF   1.1920928955078125e-7f   // np.finfo(float32).eps

// ---- launch config ----
#define BLK   256                 // 8 waves (wave32)
#define APT   4                   // anchors per thread
#define APB   (BLK*APT)           // 1024 anchors per block
#define GX    (K_DIM/APB)         // 32 blocks along K
#define NPART (GX*B_DIM)          // 1024 partial slots per accumulator
#define NACC  7                   // giou, focal_c, prop_l1, focal_pc, bce, posf, ppf

typedef __attribute__((ext_vector_type(2))) float v2f;
typedef __attribute__((ext_vector_type(8))) float v8f;

// ---- gfx1250 async global->LDS copy (ASYNCcnt path), with safe fallbacks ----
#if defined(__gfx1250__) && __has_builtin(__builtin_amdgcn_global_load_async_to_lds_b32)
#define HAVE_ASYNC_LDS 1
typedef __attribute__((address_space(1))) int gint_t;   // prints as "__device__ int *"
typedef __attribute__((address_space(3))) int lint_t;   // LDS int *
__device__ __forceinline__ void async_copy_b32(const float* gsrc, float* ldst) {
    __builtin_amdgcn_global_load_async_to_lds_b32((gint_t*)gsrc, (lint_t*)ldst, 0, 0);
}
__device__ __forceinline__ void wait_async0() {
#if __has_builtin(__builtin_amdgcn_s_wait_asynccnt)
    __builtin_amdgcn_s_wait_asynccnt(0);
#else
    asm volatile("s_wait_asynccnt 0x0" ::: "memory");
#endif
}
#else
#define HAVE_ASYNC_LDS 0
#endif

__device__ __forceinline__ float wave_sum32(float v) {
#pragma unroll
    for (int off = 16; off > 0; off >>= 1)
        v += __shfl_down(v, off, 32);
    return v;   // lane 0 holds the sum; deterministic fixed order
}

// focal term for one anchor: -alpha[t] * (1-pt)^2 * log(pt), pt = softmax(p)[t] + 1e-10
__device__ __forceinline__ float focal21(const float* __restrict__ p, int t) {
    float lg[NCLS];
    float m = -3.4e38f;
#pragma unroll
    for (int c = 0; c < NCLS; ++c) { lg[c] = p[c]; m = fmaxf(m, lg[c]); }
    float s = 0.f, en = 0.f;
#pragma unroll
    for (int c = 0; c < NCLS; ++c) {
        float e = __expf(lg[c] - m);
        s += e;
        en = (c == t) ? e : en;       // reuse the same exp term as the sum
    }
    float pt = en / s + 1e-10f;
    float av = (t == 0) ? 0.25f : 0.75f;
    float om = 1.f - pt;
    return -av * om * om * __logf(pt);
}

__global__ __launch_bounds__(BLK)
void msl_pass1(const float* __restrict__ loc,   const float* __restrict__ conf,
               const float* __restrict__ ploc,  const float* __restrict__ pconf,
               const float* __restrict__ ctr,   const float* __restrict__ priors,
               const float* __restrict__ targets, float* __restrict__ partial) {
    __shared__ float traw[NGT*3];                // raw targets row, async-staged
    __shared__ float t0s[NGT], t1s[NGT], wjs[NGT];
    __shared__ int   lbs[NGT];
    __shared__ float wred[BLK/32][NACC];

    const int tid = threadIdx.x;
    const int b   = blockIdx.y;

    // ---- stage targets[b] (96 dwords) into LDS via the async engine ----
#if HAVE_ASYNC_LDS
    if (tid < NGT*3)
        async_copy_b32(targets + b*NGT*3 + tid, &traw[tid]);
    wait_async0();                    // each issuing wave drains its own ASYNCcnt
#else
    if (tid < NGT*3) traw[tid] = targets[b*NGT*3 + tid];
#endif
    __syncthreads();                  // publish LDS to all waves

    if (tid < NGT) {
        float a = traw[tid*3 + 0];
        float e = traw[tid*3 + 1];
        float l = traw[tid*3 + 2];
        t0s[tid] = a; t1s[tid] = e;
        wjs[tid] = (e - a) * CLIPF;   // unclipped area is per-GT constant
        lbs[tid] = (int)l;
    }
    __syncthreads();

    float acc[NACC];
#pragma unroll
    for (int i = 0; i < NACC; ++i) acc[i] = 0.f;

    for (int it = 0; it < APT; ++it) {
        const int    k   = blockIdx.x*APB + it*BLK + tid;   // coalesced
        const size_t idx = (size_t)b*K_DIM + (size_t)k;
        const float* cp  = conf  + idx*NCLS;
        const float* pp  = pconf + idx*NCLS;
        if (it + 1 < APT) {                                  // gfx1250 global_prefetch_b8
            __builtin_prefetch(cp + BLK*NCLS, 0, 0);
            __builtin_prefetch(pp + BLK*NCLS, 0, 0);
        }

        const float center = priors[k];
        const float pl  = loc[idx*2 + 0],  pr  = loc[idx*2 + 1];
        const float qd0 = ploc[idx*2 + 0], qd1 = ploc[idx*2 + 1];
        const float x   = ctr[idx];

        // ---- matching: argmin over 32 GTs (first-min tiebreak, like jnp.argmin) ----
        float best = 3.4e38f; int bi = 0;
#pragma unroll
        for (int j = 0; j < NGT; ++j) {
            bool  outside = (center < t0s[j]) | (center > t1s[j]); // == (left<0)|(right<0)
            float area    = outside ? MAXNF : wjs[j];
            if (area < best) { best = area; bi = j; }
        }
        const int   conf_t = (best >= MAXNF) ? 0 : lbs[bi];
        const float tl = (center - t0s[bi]) * CLIPF;
        const float tr = (t1s[bi] - center) * CLIPF;

        // ---- IoU / GIoU of (pl,pr) vs (tl,tr) ----
        const float inter = fminf(pl, tl) + fminf(pr, tr);
        const float uni   = pl + pr + tl + tr - inter;
        const float iou   = inter / fmaxf(uni, EPSF);
        const float ac    = fmaxf(pl, tl) + fmaxf(pr, tr);
        const float giou  = iou - (ac - uni) / fmaxf(ac, EPSF);

        const int   pct  = (iou < 0.5f) ? 0 : conf_t;
        const float posf = (conf_t > 0) ? 1.f : 0.f;
        const float ppf  = (pct    > 0) ? 1.f : 0.f;

        // ---- prop-loc L1 ----
        const float pw    = pl + pr;
        const float halfw = 0.5f * pw;
        const float l1 = fabsf(qd0 - (tl - pl)/halfw) + fabsf(qd1 - (tr - pr)/halfw);

        // ---- centerness BCE on IoU of refined box ----
        const float c0  = halfw*qd0 + pl;
        const float c1  = halfw*qd1 + pr;
        const float in2 = fminf(c0, tl) + fminf(c1, tr);
        const float un2 = c0 + c1 + tl + tr - in2;
        const float iouc = fmaxf(in2 / fmaxf(un2, EPSF), 0.f);
        const float bce = fmaxf(x, 0.f) - x*iouc + log1pf(__expf(-fabsf(x)));

        acc[0] += (1.f - giou) * posf;
        acc[1] += focal21(cp, conf_t);
        acc[2] += l1 * ppf;
        acc[3] += focal21(pp, pct);
        acc[4] += bce * posf;
        acc[5] += posf;
        acc[6] += ppf;
    }

    // ---- deterministic block reduction: shfl tree then LDS ----
    const int w = tid >> 5, lane = tid & 31;
#pragma unroll
    for (int i = 0; i < NACC; ++i) {
        float s = wave_sum32(acc[i]);
        if (lane == 0) wred[w][i] = s;
    }
    __syncthreads();
    if (tid < NACC) {
        float s = 0.f;
#pragma unroll
        for (int w2 = 0; w2 < BLK/32; ++w2) s += wred[w2][tid];
        partial[tid*NPART + blockIdx.y*GX + blockIdx.x] = s;
    }
}

// Final reduction: sum 1024 f32 partials per accumulator on the matrix pipe.
// With A = all-ones (16x4), V_WMMA_F32_16X16X4_F32 gives D[m][n] = sum_k B[k][n] + C[m][n];
// packing 64 values per step into B and chaining C makes sum(row0 of D) the exact
// (reassociated) f32 total. One fully-EXEC wave per accumulator (WMMA requirement).
__global__ __launch_bounds__(256)
void msl_pass2(const float* __restrict__ partial, float* __restrict__ out) {
    __shared__ float sums[NACC];
    const int tid = threadIdx.x;
    const int w = tid >> 5, lane = tid & 31;
    if (w < NACC) {
        const float* p = partial + w*NPART;
        v2f ones; ones[0] = 1.f; ones[1] = 1.f;
        v8f d = {};
#pragma unroll
        for (int i = 0; i < NPART/64; ++i) {
            v2f bv;
            bv[0] = p[i*64 + lane];
            bv[1] = p[i*64 + 32 + lane];
            d = __builtin_amdgcn_wmma_f32_16x16x4_f32(false, ones, false, bv,
                                                      (short)0, d, false, false);
        }
        // D row 0 lives in d[0] lanes 0..15 (lanes 16..31 replicate row 8 == row 0)
        float v = (lane < 16) ? d[0] : 0.f;
        v = wave_sum32(v);
        if (lane == 0) sums[w] = v;
    }
    __syncthreads();
    if (tid == 0) {
        const float N  = fmaxf(sums[5], 1.f);
        const float PN = fmaxf(sums[6], 1.f);
        out[0] = sums[0] / N;    // loss_l
        out[1] = sums[1] / N;    // loss_c
        out[2] = sums[2] / PN;   // loss_prop_l
        out[3] = sums[3] / PN;   // loss_prop_c
        out[4] = sums[4] / N;    // loss_ct
    }
}

extern "C" void kernel_launch(void* const* d_in, const int* in_sizes, int n_in,
                              void* d_out, int out_size, void* d_ws, size_t ws_size,
                              hipStream_t stream) {
    const float* loc     = (const float*)d_in[0];   // (B,K,2)
    const float* conf    = (const float*)d_in[1];   // (B,K,21)
    const float* ploc    = (const float*)d_in[2];   // (B,K,2)
    const float* pconf   = (const float*)d_in[3];   // (B,K,21)
    const float* ctr     = (const float*)d_in[4];   // (B,K,1)
    const float* priors  = (const float*)d_in[5];   // (K,1)
    const float* targets = (const float*)d_in[6];   // (B,32,3)
    float* partial = (float*)d_ws;                  // NACC * NPART floats = 28 KB
    float* outp    = (float*)d_out;                 // 5 f32 scalars

    dim3 grid(GX, B_DIM);
    msl_pass1<<<grid, BLK, 0, stream>>>(loc, conf, ploc, pconf, ctr, priors, targets, partial);
    msl_pass2<<<1, 256, 0, stream>>>(partial, outp);
}